// DWTModelSimple_39943195853363
// MI455X (gfx1250) — compile-verified
//
#include <hip/hip_runtime.h>
#include <hip/hip_bf16.h>

// CDNA5 (gfx1250) fused 2-level Haar DWT + IDWT round trip.
// Identity-up-to-rounding, implemented as an orthogonal WMMA sandwich
// Y = M^T * (M * (X * M^T)) * M per 16x16 tile, M = blockdiag(A4 x4),
// A4 = 4x4 two-level 1-D orthonormal Haar analysis matrix.

typedef __attribute__((ext_vector_type(2))) float v2f;
typedef __attribute__((ext_vector_type(8))) float v8f;

__device__ __forceinline__ v8f wmma_f32_16x16x4(v2f a, v2f b, v8f c) {
  // 8 args: (neg_a, A, neg_b, B, c_mod, C, reuse_a, reuse_b)
  return __builtin_amdgcn_wmma_f32_16x16x4_f32(false, a, false, b, (short)0, c,
                                               false, false);
}

// Two-level 1-D orthonormal Haar on 4 samples (analysis), rows = [ss, sd, d0, d1]
__device__ __forceinline__ float haarA4(int i, int j) {
  const float H = 0.5f;
  const float R = 0.70710678118654752440f; // 1/sqrt(2)
  float r;
  if (i == 0)      r = H;
  else if (i == 1) r = (j < 2) ? H : -H;
  else if (i == 2) r = (j == 0) ? R : ((j == 1) ? -R : 0.0f);
  else             r = (j == 2) ? R : ((j == 3) ? -R : 0.0f);
  return r;
}

// Wave-internal ordering of LDS stores vs. dependent cross-lane LDS loads.
__device__ __forceinline__ void wave_lds_fence() {
  asm volatile("s_wait_dscnt 0" ::: "memory");
}

#define LDS_STRIDE 20  // 16B-aligned rows + conflict-free on 64 banks

__global__ __launch_bounds__(256) void haar_roundtrip_kernel(
    const float* __restrict__ x, float* __restrict__ y, int n_tiles) {
  const int lane = threadIdx.x & 31;
  const int wave = threadIdx.x >> 5;
  const int tile = blockIdx.x * 8 + wave;
  if (tile >= n_tiles) return;  // wave-uniform: EXEC stays all-ones

  // 512x512 plane = 32x32 tiles of 16x16
  const int plane = tile >> 10;
  const int rem   = tile & 1023;
  const int tr    = rem >> 5;
  const int tc    = rem & 31;

  const int h  = lane >> 4;   // lane half (selects K/col pair per ISA layouts)
  const int m  = lane & 15;   // A row index == B/C column index for this lane
  const int q  = m & 3;
  const int r0 = 2 * h;

  // Per-lane constant-matrix fragment values (shared by all 4 stages):
  //   u* -> A4[q][r0+i]   (M as A-frag rows / M^T as B-frag)
  //   v* -> A4[r0+i][q]   (M as B-frag / M^T as A-frag)
  const float u0 = haarA4(q, r0);
  const float u1 = haarA4(q, r0 + 1);
  const float v0 = haarA4(r0, q);
  const float v1 = haarA4(r0 + 1, q);

  __shared__ float lds[8][16 * LDS_STRIDE];
  float* t = lds[wave];  // private per-wave staging tile

  const size_t base = (size_t)plane * (512 * 512) +
                      (size_t)(tr * 16) * 512 + (size_t)(tc * 16);
  const float* src = x + base;
  float*       dst = y + base;

  // ---------- stage 1: T1 = X * M^T  (A = X from global, B = M^T const) ----
  v8f c = {};
#pragma unroll
  for (int s = 0; s < 4; ++s) {
    const float2 xa = *(const float2*)(src + (size_t)m * 512 + (4 * s + r0));
    v2f a; a.x = xa.x; a.y = xa.y;
    const bool g = ((m >> 2) == s);  // block-diagonal gate
    v2f b; b.x = g ? u0 : 0.0f; b.y = g ? u1 : 0.0f;
    c = wmma_f32_16x16x4(a, b, c);
  }
#pragma unroll
  for (int j = 0; j < 8; ++j) t[(j + 8 * h) * LDS_STRIDE + m] = c[j];
  wave_lds_fence();

  // ---------- stage 2: T2 = M * T1  (A = M const, B = T1 from LDS) ---------
  v8f c2 = {};
#pragma unroll
  for (int s = 0; s < 4; ++s) {
    const bool g = ((m >> 2) == s);
    v2f a; a.x = g ? u0 : 0.0f; a.y = g ? u1 : 0.0f;
    v2f b;
    b.x = t[(4 * s + r0) * LDS_STRIDE + m];
    b.y = t[(4 * s + r0 + 1) * LDS_STRIDE + m];
    c2 = wmma_f32_16x16x4(a, b, c2);
  }
  wave_lds_fence();  // all reads of T1 done before overwrite
#pragma unroll
  for (int j = 0; j < 8; ++j) t[(j + 8 * h) * LDS_STRIDE + m] = c2[j];
  wave_lds_fence();

  // ---------- stage 3: T3 = T2 * M  (A = T2 from LDS, B = M const) ---------
  v8f c3 = {};
#pragma unroll
  for (int s = 0; s < 4; ++s) {
    const float2 ta = *(const float2*)&t[m * LDS_STRIDE + (4 * s + r0)];
    v2f a; a.x = ta.x; a.y = ta.y;
    const bool g = ((m >> 2) == s);
    v2f b; b.x = g ? v0 : 0.0f; b.y = g ? v1 : 0.0f;
    c3 = wmma_f32_16x16x4(a, b, c3);
  }
  wave_lds_fence();
#pragma unroll
  for (int j = 0; j < 8; ++j) t[(j + 8 * h) * LDS_STRIDE + m] = c3[j];
  wave_lds_fence();

  // ---------- stage 4: Y = M^T * T3  (A = M^T const, B = T3 from LDS) ------
  v8f c4 = {};
#pragma unroll
  for (int s = 0; s < 4; ++s) {
    const bool g = ((m >> 2) == s);
    v2f a; a.x = g ? v0 : 0.0f; a.y = g ? v1 : 0.0f;
    v2f b;
    b.x = t[(4 * s + r0) * LDS_STRIDE + m];
    b.y = t[(4 * s + r0 + 1) * LDS_STRIDE + m];
    c4 = wmma_f32_16x16x4(a, b, c4);
  }

  // C/D layout: VGPR j -> row (j + 8*h), column m. 64B-contiguous per half.
#pragma unroll
  for (int j = 0; j < 8; ++j)
    dst[(size_t)(j + 8 * h) * 512 + m] = c4[j];
}

extern "C" void kernel_launch(void* const* d_in, const int* in_sizes, int n_in,
                              void* d_out, int out_size, void* d_ws,
                              size_t ws_size, hipStream_t stream) {
  const float* x = (const float*)d_in[0];
  float* y = (float*)d_out;

  const int total  = in_sizes[0];          // B*C*512*512
  const int planes = total / (512 * 512);  // 96
  const int n_tiles = planes * 32 * 32;    // 16x16 tiles
  const int blocks  = (n_tiles + 7) / 8;   // 8 waves (tiles) per 256-thread block

  haar_roundtrip_kernel<<<blocks, 256, 0, stream>>>(x, y, n_tiles);
}